// VarianceSmoothLoss_46179488367045
// MI455X (gfx1250) — compile-verified
//
#include <hip/hip_runtime.h>
#include <hip/hip_bf16.h>

typedef __attribute__((ext_vector_type(2))) float v2f;
typedef __attribute__((ext_vector_type(8))) float v8f;

#define NSEG 64
#define TILE 256           // rows per block-tile (8 waves x 32 rows)
#define WAVES 8
#define MAIN_BLOCKS 512

// Workspace layout (floats): ws[0..63] = counts, ws[64 + seg*16 + c]:
//   c in 0..7 = sum_d, c in 8..15 = sumsq_d.  Total 1088 floats.

__global__ void vsl_init(float* __restrict__ ws) {
  int i = threadIdx.x + blockIdx.x * blockDim.x;
  for (int k = i; k < NSEG + NSEG * 16; k += blockDim.x * gridDim.x) ws[k] = 0.0f;
}

// Issue async copies of one 32B row + its label directly into LDS (GVS form).
#define ISSUE_ASYNC(tile_, ldsx_, ldsl_)                                        \
  do {                                                                          \
    unsigned r_  = (unsigned)((tile_)*TILE + tid);                              \
    unsigned vx_ = r_ * 32u;                                                    \
    unsigned vl_ = r_ * 4u;                                                     \
    asm volatile("global_load_async_to_lds_b128 %0, %1, %2\n\t"                 \
                 "global_load_async_to_lds_b128 %0, %1, %2 offset:16"           \
                 :: "v"(ldsx_), "v"(vx_), "s"(xs64) : "memory");                \
    asm volatile("global_load_async_to_lds_b32 %0, %1, %2"                      \
                 :: "v"(ldsl_), "v"(vl_), "s"(ls64) : "memory");                \
  } while (0)

__global__ __launch_bounds__(256) void vsl_main(const float* __restrict__ x,
                                                const int* __restrict__ lab,
                                                float* __restrict__ ws, int n) {
  __shared__ float    xt[2][WAVES][32][8];  // double-buffered per-wave rows
  __shared__ int      lt[2][WAVES][32];     // double-buffered per-wave labels
  __shared__ float    accS[NSEG][16];       // block-level C accumulator
  __shared__ unsigned cnt[NSEG];            // block-level counts

  const int tid  = threadIdx.x;
  const int w    = tid >> 5;
  const int lane = tid & 31;
  const int m    = lane & 15;       // M (A) / N (B,C) index
  const int half = lane >> 4;       // selects K pair {0,1} vs {2,3}
  const int kb   = half << 1;
  const int dsel = m & 7;           // which of the 8 dims feeds this B column
  const bool sq  = (m & 8) != 0;    // cols 8..15 carry x^2

  const unsigned long long xs64 = (unsigned long long)x;
  const unsigned long long ls64 = (unsigned long long)lab;
  const unsigned ldsX0 = (unsigned)(size_t)&xt[0][w][lane][0];
  const unsigned ldsX1 = (unsigned)(size_t)&xt[1][w][lane][0];
  const unsigned ldsL0 = (unsigned)(size_t)&lt[0][w][lane];
  const unsigned ldsL1 = (unsigned)(size_t)&lt[1][w][lane];

  for (int i2 = tid; i2 < NSEG * 16; i2 += 256) (&accS[0][0])[i2] = 0.0f;
  if (tid < NSEG) cnt[tid] = 0u;
  __syncthreads();

  v8f cacc[4] = {};   // 4 segment tiles of 16 x 16 f32 accumulators

  const int numTiles = n / TILE;
  const int stride   = (int)gridDim.x;
  int t = blockIdx.x;
  if (t < numTiles) ISSUE_ASYNC(t, ldsX0, ldsL0);   // prologue (block-uniform)

  int buf = 0;
  for (; t < numTiles; t += stride) {
    const int nxt = t + stride;
    const unsigned ldsXn = buf ? ldsX0 : ldsX1;
    const unsigned ldsLn = buf ? ldsL0 : ldsL1;
    if (nxt < numTiles) {                       // uniform per block
      ISSUE_ASYNC(nxt, ldsXn, ldsLn);
      if (nxt + stride < numTiles)              // warm L2 two tiles ahead
        __builtin_prefetch(x + (size_t)(nxt + stride) * (TILE * 8) + (size_t)tid * 8, 0, 1);
      asm volatile("s_wait_asynccnt 0x3" ::: "memory");  // current buffer done
    } else {
      asm volatile("s_wait_asynccnt 0x0" ::: "memory");
    }

    const float (*xb)[8] = xt[buf][w];
    const int*   lbp     = lt[buf][w];

    int L = lbp[lane];
    atomicAdd(&cnt[L], 1u);                     // ds_add_u32 histogram

    // 8 groups of 4 rows; per group: 1 B build + 4 x v_wmma_f32_16x16x4_f32
#pragma unroll
    for (int g = 0; g < 8; ++g) {
      const int rb = (g << 2) + kb;
      const int L0 = lbp[rb];
      const int L1 = lbp[rb + 1];
      float b0 = xb[rb][dsel];
      float b1 = xb[rb + 1][dsel];
      b0 = sq ? b0 * b0 : b0;                   // cndmask, EXEC unchanged
      b1 = sq ? b1 * b1 : b1;
      v2f B;
      B[0] = b0;
      B[1] = b1;
#pragma unroll
      for (int t4 = 0; t4 < 4; ++t4) {
        const int base = t4 << 4;
        v2f A;                                  // one-hot of this row's label
        A[0] = (L0 == base + m) ? 1.0f : 0.0f;
        A[1] = (L1 == base + m) ? 1.0f : 0.0f;
        cacc[t4] = __builtin_amdgcn_wmma_f32_16x16x4_f32(
            false, A, false, B, (short)0, cacc[t4], false, false);
      }
    }
    buf ^= 1;
  }

  // Fold per-wave C accumulators into block LDS accumulator.
#pragma unroll
  for (int t4 = 0; t4 < 4; ++t4) {
#pragma unroll
    for (int v = 0; v < 8; ++v) {
      atomicAdd(&accS[(t4 << 4) + v + (half << 3)][m], cacc[t4][v]);
    }
  }
  __syncthreads();

  // Block -> global (f32 atomics, 1088 per block)
  for (int i2 = tid; i2 < NSEG; i2 += 256)
    atomicAdd(&ws[i2], (float)cnt[i2]);
  for (int i2 = tid; i2 < NSEG * 16; i2 += 256)
    atomicAdd(&ws[NSEG + i2], (&accS[0][0])[i2]);

  // Scalar tail (n % 256 rows) — empty for N = 4,000,000.
  if (blockIdx.x == 0) {
    int r = (n / TILE) * TILE + tid;
    if (r < n) {
      int L = lab[r];
      atomicAdd(&ws[L], 1.0f);
      for (int d2 = 0; d2 < 8; ++d2) {
        float xv = x[(size_t)r * 8 + d2];
        atomicAdd(&ws[NSEG + L * 16 + d2], xv);
        atomicAdd(&ws[NSEG + L * 16 + 8 + d2], xv * xv);
      }
    }
  }
}

__global__ void vsl_final(const float* __restrict__ ws, float* __restrict__ out) {
  __shared__ float red[64];
  int s = threadIdx.x;   // 64 threads, one per segment
  float c = ws[s];
  float val = 0.0f;
  if (s != 0 && c > 0.0f) {
    float acc = 0.0f;
#pragma unroll
    for (int d2 = 0; d2 < 8; ++d2) {
      float S  = ws[64 + s * 16 + d2];
      float SS = ws[64 + s * 16 + 8 + d2];
      acc += SS - S * S / c;                // sum (x - mean)^2 identity
    }
    val = acc / (c * 8.0f);
  }
  red[s] = val;
  __syncthreads();
  if (s == 0) {
    float tsum = 0.0f;
    for (int i = 0; i < 64; ++i) tsum += red[i];
    out[0] = tsum;
  }
}

extern "C" void kernel_launch(void* const* d_in, const int* in_sizes, int n_in,
                              void* d_out, int out_size, void* d_ws, size_t ws_size,
                              hipStream_t stream) {
  const float* x  = (const float*)d_in[0];
  const int* lab  = (const int*)d_in[1];
  float* out      = (float*)d_out;
  float* ws       = (float*)d_ws;
  const int n     = in_sizes[1];   // number of rows (labels count)

  vsl_init<<<4, 256, 0, stream>>>(ws);
  vsl_main<<<MAIN_BLOCKS, 256, 0, stream>>>(x, lab, ws, n);
  vsl_final<<<1, 64, 0, stream>>>(ws, out);
}